// multilayer_11897059410367
// MI455X (gfx1250) — compile-verified
//
#include <hip/hip_runtime.h>

// ---------------------------------------------------------------------------
// Light-field multiplane composition for MI455X (gfx1250).
//
// grid = identity + planes[n,l]*shift[a]  ==> constant sub-pixel translation
// (|dx|,|dy| < 2 px) with border clamp. Per (n, view, 32x32 tile): stage the
// layer data into LDS with the Tensor Data Mover, software-pipelined:
// 6 stages (3 layers x 2 channel-groups of 6), double-buffered in LDS with
// two TDM loads in flight (s_wait_tensorcnt(1) steady state / (0) drain),
// so DMA overlaps the separable bilinear blend + multiplicative composition.
// Thread = 4 consecutive columns of one row -> shared wx, 10 LDS taps per
// channel (2 rows x 5 cols), float4 output stores.
// ---------------------------------------------------------------------------

#define LF_H   256
#define LF_W   384
#define LF_L   3
#define LF_RC  12            // rank(4) * C(3)
#define LF_C   3
#define LF_A2  49
#define LF_HW  (LF_H * LF_W)
#define TILE   32
#define HALO   36            // 32 + 2*2 halo (shift magnitude < 2 px)
#define HALO2  (HALO * HALO)
#define GRP    6             // channels per pipeline stage
#define NSTAGE (LF_L * 2)    // 6 stages
#define GRPF   (GRP * HALO2) // floats per LDS buffer (7776)
#define GRPB   (GRPF * 4)    // bytes per LDS buffer (31104)

typedef unsigned int       u32;
typedef unsigned long long u64;
typedef __attribute__((ext_vector_type(4))) u32 tdm_v4u;
typedef __attribute__((ext_vector_type(8))) int tdm_v8i;
typedef __attribute__((ext_vector_type(4))) int tdm_v4i;

#if __has_builtin(__builtin_amdgcn_tensor_load_to_lds)
#define LF_HAVE_TDM 1
#endif

__device__ __forceinline__ void lf_wait_tensorcnt0() {
#if __has_builtin(__builtin_amdgcn_s_wait_tensorcnt)
  __builtin_amdgcn_s_wait_tensorcnt(0);
#else
  asm volatile("s_wait_tensorcnt 0" ::: "memory");
#endif
}
__device__ __forceinline__ void lf_wait_tensorcnt1() {
#if __has_builtin(__builtin_amdgcn_s_wait_tensorcnt)
  __builtin_amdgcn_s_wait_tensorcnt(1);
#else
  asm volatile("s_wait_tensorcnt 1" ::: "memory");
#endif
}

#ifdef LF_HAVE_TDM
// One TDM load of a 3-D tile: x = HALO floats, y = HALO rows (stride W),
// z = GRP channel planes (stride H*W). D# layout per CDNA5 ISA section 8.
__device__ __forceinline__ void lf_tdm_load_group(u32 lds_byte_addr, u64 gaddr) {
  tdm_v4u g0;
  g0.x = 1u;                                                // count=1, user mode
  g0.y = lds_byte_addr;                                     // lds_addr
  g0.z = (u32)(gaddr & 0xFFFFFFFFull);                      // global_addr[31:0]
  g0.w = (u32)((gaddr >> 32) & 0x01FFFFFFull) | (2u << 30); // addr[56:32] | type=2

  tdm_v8i g1;
  g1[0] = (int)(2u << 16);                                  // data_size = 4 bytes
  g1[1] = (int)((u32)HALO << 16);                           // tensor_dim0[15:0]
  g1[2] = (int)((u32)HALO << 16);                           // dim0 hi=0 | tensor_dim1[15:0]
  g1[3] = (int)((u32)HALO << 16);                           // dim1 hi=0 | tile_dim0
  g1[4] = (int)((u32)HALO | ((u32)GRP << 16));              // tile_dim1 | tile_dim2
  g1[5] = (int)(u32)LF_W;                                   // tensor_dim0_stride lo
  g1[6] = (int)(((u32)LF_HW & 0xFFFFu) << 16);              // stride0 hi=0 | stride1[15:0]
  g1[7] = (int)((u32)LF_HW >> 16);                          // stride1[47:16]

  tdm_v4i g2 = {GRP, 0, 0, 0};                              // tensor_dim2 = 6
  tdm_v4i g3 = {0, 0, 0, 0};

#if __has_include(<hip/amd_detail/amd_gfx1250_TDM.h>)
  tdm_v8i g4 = {0, 0, 0, 0, 0, 0, 0, 0};                    // clang-23 6-arg form
  __builtin_amdgcn_tensor_load_to_lds(g0, g1, g2, g3, g4, 0);
#else
  __builtin_amdgcn_tensor_load_to_lds(g0, g1, g2, g3, 0);   // ROCm 7.2 5-arg form
#endif
}
#endif

// Blend 6 channels of one stage into acc[][G*6 .. G*6+5] (compile-time G).
template <int G>
__device__ __forceinline__ void lf_compute_group(
    const float* __restrict__ b, float (&acc)[4][LF_RC],
    int r0, int r1, float wy, float wy0,
    int w0, float dx, int gx_base) {
  const float xq = (float)w0 + dx;
  if ((xq >= 0.0f) && (xq + 3.0f < (float)(LF_W - 1))) {
    // fast path: no x-clamp in this quad; taps = 2 rows x 5 consecutive cols
    const float x0f = floorf(xq);
    const float wx  = xq - x0f;
    const float wx0 = 1.0f - wx;
    const int   lxb = (int)x0f - gx_base;                   // in [0, 31]
    const float* p0 = b + r0 + lxb;
    const float* p1 = b + r1 + lxb;
#pragma unroll
    for (int cc = 0; cc < GRP; ++cc) {
      const int off = cc * HALO2;                           // DS immediate
      float V[5];
#pragma unroll
      for (int j = 0; j < 5; ++j)
        V[j] = fmaf(wy, p1[off + j], wy0 * p0[off + j]);    // vertical blend
#pragma unroll
      for (int i = 0; i < 4; ++i)
        acc[i][G * GRP + cc] *= fmaf(wx, V[i + 1], wx0 * V[i]);
    }
  } else {
    // generic path: per-pixel clamped taps (image-border quads only)
#pragma unroll
    for (int i = 0; i < 4; ++i) {
      const float x   = fminf(fmaxf((float)(w0 + i) + dx, 0.0f),
                              (float)(LF_W - 1));
      const float x0f = floorf(x);
      const int   x0  = (int)x0f;
      const int   x1  = min(x0 + 1, LF_W - 1);
      const float wx  = x - x0f;
      const float wx0 = 1.0f - wx;
      const int   lx0 = x0 - gx_base;
      const int   lx1 = x1 - gx_base;
      const float w00 = wx0 * wy0, w01 = wx * wy0;
      const float w10 = wx0 * wy,  w11 = wx * wy;
#pragma unroll
      for (int cc = 0; cc < GRP; ++cc) {
        const int off = cc * HALO2;
        const float v = w00 * b[off + r0 + lx0] + w01 * b[off + r0 + lx1]
                      + w10 * b[off + r1 + lx0] + w11 * b[off + r1 + lx1];
        acc[i][G * GRP + cc] *= v;
      }
    }
  }
}

__global__ __launch_bounds__(256) void lf_mpi_kernel(
    const float* __restrict__ low_rank,   // (N, L, R, C, H, W)
    const float* __restrict__ planes,     // (N, L)
    float* __restrict__ out) {            // (N, A2, C, H, W)
  __shared__ float tile[2 * GRPF];        // 62208 bytes, double-buffered

  const int tilesX = LF_W / TILE;                           // 12
  const int tIdx = blockIdx.x;                              // 0..95
  const int a    = blockIdx.y;                              // 0..48
  const int n    = blockIdx.z;                              // 0..1
  const int tx0  = (tIdx % tilesX) * TILE;
  const int ty0  = (tIdx / tilesX) * TILE;

  // shifts: idx = [-3..3]; a = k*7 + l (k: y, l: x)
  const float llf = (float)((a % 7) - 3);
  const float kkf = (float)((a / 7) - 3);

  // thread = 4 consecutive columns of one row
  const int q   = (int)(threadIdx.x & 7);
  const int row = (int)(threadIdx.x >> 3);
  const int w0  = tx0 + q * 4;
  const int h   = ty0 + row;

  // LDS window clamped inside the image so clamped coords stay in-window.
  const int gx_base = min(max(tx0 - 2, 0), LF_W - HALO);
  const int gy_base = min(max(ty0 - 2, 0), LF_H - HALO);

  float acc[4][LF_RC];
#pragma unroll
  for (int i = 0; i < 4; ++i)
#pragma unroll
    for (int ch = 0; ch < LF_RC; ++ch) acc[i][ch] = 1.0f;

  const u32 lds0 = (u32)(u64)(const void*)&tile[0];         // low 32b = LDS offset

  // stage s: layer l = s>>1, channel group g = s&1 (6 planes each)
  auto stage_src = [&](int s) -> const float* {
    const int l = s >> 1, g = s & 1;
    return low_rank + (u64)((n * LF_L + l) * LF_RC + g * GRP) * LF_HW
                    + (u64)gy_base * LF_W + (u64)gx_base;
  };

#ifdef LF_HAVE_TDM
  const bool issuer =
      (__builtin_amdgcn_readfirstlane((int)(threadIdx.x >> 5)) == 0);
  if (issuer)
    lf_tdm_load_group(lds0, (u64)(const void*)stage_src(0)); // prime stage 0
#endif

#pragma unroll 1
  for (int s = 0; s < NSTAGE; ++s) {
#ifdef LF_HAVE_TDM
    if (issuer) {
      if (s + 1 < NSTAGE) {
        // buf[(s+1)&1] was last read at stage s-1, released by its barrier.
        lf_tdm_load_group(lds0 + (u32)((s + 1) & 1) * (u32)GRPB,
                          (u64)(const void*)stage_src(s + 1));
        lf_wait_tensorcnt1();   // in-order TDM: <=1 outstanding => stage s done
      } else {
        lf_wait_tensorcnt0();   // drain last stage
      }
    }
#else
    {
      const float* srcS = stage_src(s);
      float* dst = tile + (u32)(s & 1) * GRPF;
      for (int idx = (int)threadIdx.x; idx < GRPF; idx += 256) {
        const int cc  = idx / HALO2;
        const int rem = idx - cc * HALO2;
        const int ry  = rem / HALO;
        const int rx  = rem - ry * HALO;
        dst[idx] = srcS[(u64)cc * LF_HW + (u64)ry * LF_W + rx];
      }
    }
#endif
    __syncthreads();            // stage s data visible to all waves

    const int   l  = s >> 1;
    const float pl = planes[n * LF_L + l];
    const float dx = pl * (-1.0f / (float)LF_W) * llf * 0.5f * (float)(LF_W - 1);
    const float dy = pl * (-1.0f / (float)LF_H) * kkf * 0.5f * (float)(LF_H - 1);

    // y taps: one row per thread
    const float y   = fminf(fmaxf((float)h + dy, 0.0f), (float)(LF_H - 1));
    const float y0f = floorf(y);
    const int   y0  = (int)y0f;
    const int   y1  = min(y0 + 1, LF_H - 1);
    const float wy  = y - y0f;
    const float wy0 = 1.0f - wy;
    const int   r0  = (y0 - gy_base) * HALO;
    const int   r1  = (y1 - gy_base) * HALO;

    const float* b = tile + (u32)(s & 1) * GRPF;
    if ((s & 1) == 0)
      lf_compute_group<0>(b, acc, r0, r1, wy, wy0, w0, dx, gx_base);
    else
      lf_compute_group<1>(b, acc, r0, r1, wy, wy0, w0, dx, gx_base);

    __syncthreads();            // reads done; buffer reusable at stage s+2
  }

  // mean over rank (rc = r*3 + c); float4 store per channel, coalesced over w.
  const u64 outBase = (u64)(n * LF_A2 + a) * LF_C * LF_HW
                    + (u64)h * LF_W + (u64)w0;
#pragma unroll
  for (int c = 0; c < LF_C; ++c) {
    float4 v;
    v.x = 0.25f * (acc[0][c] + acc[0][3 + c] + acc[0][6 + c] + acc[0][9 + c]);
    v.y = 0.25f * (acc[1][c] + acc[1][3 + c] + acc[1][6 + c] + acc[1][9 + c]);
    v.z = 0.25f * (acc[2][c] + acc[2][3 + c] + acc[2][6 + c] + acc[2][9 + c]);
    v.w = 0.25f * (acc[3][c] + acc[3][3 + c] + acc[3][6 + c] + acc[3][9 + c]);
    *reinterpret_cast<float4*>(out + outBase + (u64)c * LF_HW) = v;
  }
}

extern "C" void kernel_launch(void* const* d_in, const int* in_sizes, int n_in,
                              void* d_out, int out_size, void* d_ws, size_t ws_size,
                              hipStream_t stream) {
  (void)in_sizes; (void)n_in; (void)d_ws; (void)ws_size; (void)out_size;
  const float* low_rank = (const float*)d_in[0];   // (2,3,4,3,256,384) f32
  const float* planes   = (const float*)d_in[1];   // (2,3) f32
  float* out            = (float*)d_out;           // (2,49,3,256,384) f32

  dim3 grid((LF_W / TILE) * (LF_H / TILE), LF_A2, 2);  // (96, 49, 2)
  dim3 block(256);
  lf_mpi_kernel<<<grid, block, 0, stream>>>(low_rank, planes, out);
}